// Traveltime_47150150975559
// MI455X (gfx1250) — compile-verified
//
#include <hip/hip_runtime.h>
#include <stdint.h>

// ---------------------------------------------------------------------------
// Reference reduces to: per-column argmax of a^2 (fft∘ifft == identity), then
// mean of squared index differences. Memory-bound: 134 MB single pass.
// ---------------------------------------------------------------------------

typedef float v2f __attribute__((ext_vector_type(2)));
typedef float v8f __attribute__((ext_vector_type(8)));

constexpr int NT  = 4096;   // time samples (argmax axis)
constexpr int NTR = 4096;   // traces (columns)
constexpr int ROWS_PER_CHUNK = 64;   // rows scanned per block (grid.y = 64)
constexpr int P1_BLOCK = 128;        // threads per block, 4 columns each

// Pack (value, index) so that u64 atomicMax == argmax with first-max-wins.
// value = a^2 >= 0, so its IEEE bit pattern is monotonic in value.
// ~idx in the low word: equal values -> larger ~idx -> smaller idx wins.
__device__ __forceinline__ unsigned long long pack_key(float v, int idx) {
    return ((unsigned long long)__float_as_uint(v) << 32) |
           (unsigned long long)(unsigned int)(~idx);
}

__global__ void init_ws_kernel(unsigned long long* __restrict__ ws, int n) {
    int i = blockIdx.x * blockDim.x + threadIdx.x;
    if (i < n) ws[i] = 0ull;
}

// Streaming argmax pass: each thread owns 4 consecutive columns (float4 load
// per row per array -> coalesced global_load_b128 across the wave).
__global__ void __launch_bounds__(P1_BLOCK)
argmax_pass_kernel(const float* __restrict__ x,
                   const float* __restrict__ y,
                   unsigned long long* __restrict__ wsx,
                   unsigned long long* __restrict__ wsy) {
    const int col4 = blockIdx.x * P1_BLOCK + threadIdx.x;  // float4-column index
    const int r0   = blockIdx.y * ROWS_PER_CHUNK;
    const int ld4  = NTR / 4;

    const float4* __restrict__ xv = (const float4*)x;
    const float4* __restrict__ yv = (const float4*)y;

    float bx0 = -1.f, bx1 = -1.f, bx2 = -1.f, bx3 = -1.f;
    float by0 = -1.f, by1 = -1.f, by2 = -1.f, by3 = -1.f;
    int   ix0 = 0, ix1 = 0, ix2 = 0, ix3 = 0;
    int   iy0 = 0, iy1 = 0, iy2 = 0, iy3 = 0;

    #pragma unroll 4
    for (int r = r0; r < r0 + ROWS_PER_CHUNK; ++r) {
        const float4 a = xv[(size_t)r * ld4 + col4];
        const float4 b = yv[(size_t)r * ld4 + col4];
        float s;
        s = a.x * a.x; if (s > bx0) { bx0 = s; ix0 = r; }
        s = a.y * a.y; if (s > bx1) { bx1 = s; ix1 = r; }
        s = a.z * a.z; if (s > bx2) { bx2 = s; ix2 = r; }
        s = a.w * a.w; if (s > bx3) { bx3 = s; ix3 = r; }
        s = b.x * b.x; if (s > by0) { by0 = s; iy0 = r; }
        s = b.y * b.y; if (s > by1) { by1 = s; iy1 = r; }
        s = b.z * b.z; if (s > by2) { by2 = s; iy2 = r; }
        s = b.w * b.w; if (s > by3) { by3 = s; iy3 = r; }
    }

    const int c = col4 * 4;
    atomicMax(&wsx[c + 0], pack_key(bx0, ix0));
    atomicMax(&wsx[c + 1], pack_key(bx1, ix1));
    atomicMax(&wsx[c + 2], pack_key(bx2, ix2));
    atomicMax(&wsx[c + 3], pack_key(bx3, ix3));
    atomicMax(&wsy[c + 0], pack_key(by0, iy0));
    atomicMax(&wsy[c + 1], pack_key(by1, iy1));
    atomicMax(&wsy[c + 2], pack_key(by2, iy2));
    atomicMax(&wsy[c + 3], pack_key(by3, iy3));
}

// Finalize: unpack argmax indices, squared diff per column, reduce 256 thread
// partials with V_WMMA_F32_16X16X4_F32 (B = all-ones => D row-sums everything;
// any bijection of the 64 values into the A fragment is valid for a total sum).
__global__ void __launch_bounds__(256)
finalize_kernel(const unsigned long long* __restrict__ wsx,
                const unsigned long long* __restrict__ wsy,
                float* __restrict__ out) {
    __shared__ float part[256];
    const int t = threadIdx.x;

    float acc = 0.f;
    #pragma unroll
    for (int k = 0; k < NTR; k += 256) {
        const unsigned long long px = wsx[k + t];
        const unsigned long long py = wsy[k + t];
        const int ix = (int)(~(unsigned int)(px & 0xFFFFFFFFull));
        const int iy = (int)(~(unsigned int)(py & 0xFFFFFFFFull));
        const float d = (float)(ix - iy);
        acc += d * d;
    }
    part[t] = acc;
    __syncthreads();

    if (t < 32) {                    // wave 0 only: EXEC all-ones for WMMA
        v8f cacc = {0.f, 0.f, 0.f, 0.f, 0.f, 0.f, 0.f, 0.f};
        v2f ones;
        ones[0] = 1.f;
        ones[1] = 1.f;
        #pragma unroll
        for (int it = 0; it < 4; ++it) {
            v2f a;                  // 64 distinct values per WMMA (32 lanes x 2 VGPRs)
            a[0] = part[it * 64 + t * 2 + 0];
            a[1] = part[it * 64 + t * 2 + 1];
            // D = A x ones + C  => after 4 iters, sum_m D[m][0] == sum of all 256
            cacc = __builtin_amdgcn_wmma_f32_16x16x4_f32(
                false, a, false, ones, (short)0, cacc, false, false);
        }
        float s = cacc[0] + cacc[1] + cacc[2] + cacc[3] +
                  cacc[4] + cacc[5] + cacc[6] + cacc[7];   // rows 0-7 (lane 0) / 8-15 (lane 16)
        const float s_hi = __shfl(s, 16, 32);
        if (t == 0) out[0] = (s + s_hi) / (float)NTR;
    }
}

extern "C" void kernel_launch(void* const* d_in, const int* in_sizes, int n_in,
                              void* d_out, int out_size, void* d_ws, size_t ws_size,
                              hipStream_t stream) {
    (void)in_sizes; (void)n_in; (void)out_size; (void)ws_size;

    const float* x = (const float*)d_in[0];
    const float* y = (const float*)d_in[1];
    float* out = (float*)d_out;

    unsigned long long* wsx = (unsigned long long*)d_ws;   // 4096 packed keys
    unsigned long long* wsy = wsx + NTR;                   // 4096 packed keys

    // Zero packed keys every launch (graph-replay safe).
    init_ws_kernel<<<(2 * NTR + 255) / 256, 256, 0, stream>>>(wsx, 2 * NTR);

    // Single streaming pass over both arrays: 512 blocks.
    dim3 grid(NTR / (P1_BLOCK * 4), NT / ROWS_PER_CHUNK);
    argmax_pass_kernel<<<grid, P1_BLOCK, 0, stream>>>(x, y, wsx, wsy);

    // Tiny finalize (64 KB of keys -> scalar), WMMA f32 reduction.
    finalize_kernel<<<1, 256, 0, stream>>>(wsx, wsy, out);
}